// NEmbedding_25915832664242
// MI455X (gfx1250) — compile-verified
//
#include <hip/hip_runtime.h>

typedef float v2f __attribute__((ext_vector_type(2)));
typedef float v8f __attribute__((ext_vector_type(8)));

#define B_  32768
#define F_  128
#define NB_ 16
#define E_  32

// One wave per (feature f, 16-row batch tile). PLE boundaries are fetched
// per-lane with `hi` folded into the address (no register-array selects),
// then D[16,32] = enc[16,16] @ W[f][16,32] via 2 N-tiles x 4 chained
// V_WMMA_F32_16X16X4_F32, + bias, ReLU, store.
__global__ __launch_bounds__(256) void ple_embedding_wmma_kernel(
    const float* __restrict__ x,       // [B, F]
    const float* __restrict__ bounds,  // [F, NB]
    const float* __restrict__ W,       // [F, NB, E]
    const float* __restrict__ bias,    // [F, E]
    float* __restrict__ out)           // [B, F, E]
{
  const int wave = threadIdx.x >> 5;
  const int lane = threadIdx.x & 31;
  const int n    = lane & 15;   // A-row M / B,C,D column N
  const int hi   = lane >> 4;   // half-wave select

  const int fgroup = blockIdx.x & 15;   // 16 groups of 8 consecutive features
  const int mtile  = blockIdx.x >> 4;   // 2048 batch tiles of 16 rows
  const int f  = fgroup * 8 + wave;
  const int m0 = mtile * 16;

  const float xv = x[(size_t)(m0 + n) * F_ + f];   // x fits in L2 (16.8 MB)
  const float* bp = bounds + f * NB_;

  // ---- A fragments (16x4 f32 ISA layout): K-step ks covers K=4ks..4ks+3.
  // Lane holds bins {base, base+1} with base = 4ks + 2*hi; boundary values
  // come straight from (cache-resident) loads with hi folded into the
  // address, so no dynamic register indexing / select trees can appear.
  v2f afrag[4];
#pragma unroll
  for (int ks = 0; ks < 4; ++ks) {
    const int base = 4 * ks + 2 * hi;
    const v2f lo2 = *(const v2f*)(bp + base);         // bnd[base], bnd[base+1]
    // h(t=1) = bnd[base+2]; for the j==15 lane (ks==3 && hi) clamp the load
    // to index 15 (stays in-bounds) and override with bext's default -1.0.
    const int hidx = (ks == 3) ? (14 + hi) : (base + 2);
    const float hb = bp[hidx];
    const float h1 = (ks == 3 && hi) ? -1.0f : hb;

    // t = 0 : bin j = base   (h = bnd[base+1] = lo2.y, j<15 always true)
    {
      const bool jgt0  = (ks > 0) || (hi != 0);       // j>0
      const bool left  = jgt0 && (xv < lo2.x);
      const bool right = (xv >= lo2.y);
      const float v = (xv - lo2.x) * __builtin_amdgcn_rcpf(lo2.y - lo2.x);
      afrag[ks].x = left ? 0.0f : (right ? 1.0f : v);
    }
    // t = 1 : bin j = base+1 (j>0 always true)
    {
      const bool jlt15 = (ks < 3) || (hi == 0);       // j<15
      const bool left  = (xv < lo2.y);
      const bool right = jlt15 && (xv >= h1);
      const float v = (xv - lo2.y) * __builtin_amdgcn_rcpf(h1 - lo2.y);
      afrag[ks].y = left ? 0.0f : (right ? 1.0f : v);
    }
  }

  const float* Wf = W    + (size_t)f * (NB_ * E_);
  const float* bf = bias + (size_t)f * E_;
  float* of       = out  + (size_t)f * E_;

#pragma unroll
  for (int nt = 0; nt < 2; ++nt) {   // E=32 -> two 16-wide N tiles
    v8f acc = {};
#pragma unroll
    for (int ks = 0; ks < 4; ++ks) {
      const int k0 = 4 * ks + 2 * hi;  // B 4x16 layout mirrors A layout
      v2f bfrag;
      bfrag.x = Wf[(size_t)(k0 + 0) * E_ + nt * 16 + n];  // 64B-contig per half-wave
      bfrag.y = Wf[(size_t)(k0 + 1) * E_ + nt * 16 + n];
      // D = A x B + C  (fp32, chained accumulation over K)
      acc = __builtin_amdgcn_wmma_f32_16x16x4_f32(
          /*neg_a=*/false, afrag[ks], /*neg_b=*/false, bfrag,
          /*c_mod=*/(short)0, acc, /*reuse_a=*/false, /*reuse_b=*/false);
    }
    const float bv = bf[nt * 16 + n];  // bias for column N (uniform over M)
#pragma unroll
    for (int j = 0; j < 8; ++j) {      // C/D layout: M = j + 8*hi, N = n
      float d = acc[j] + bv;
      d = d > 0.0f ? d : 0.0f;         // ReLU
      const int M = j + 8 * hi;
      of[(size_t)(m0 + M) * (F_ * E_) + nt * 16 + n] = d;
    }
  }
}

extern "C" void kernel_launch(void* const* d_in, const int* in_sizes, int n_in,
                              void* d_out, int out_size, void* d_ws, size_t ws_size,
                              hipStream_t stream) {
  (void)in_sizes; (void)n_in; (void)out_size; (void)d_ws; (void)ws_size;
  const float* x      = (const float*)d_in[0];
  const float* bounds = (const float*)d_in[1];
  const float* W      = (const float*)d_in[2];
  const float* b      = (const float*)d_in[3];
  float* out = (float*)d_out;

  const int grid = (F_ / 8) * (B_ / 16);  // 16 f-groups * 2048 batch tiles = 32768
  ple_embedding_wmma_kernel<<<grid, 256, 0, stream>>>(x, bounds, W, b, out);
}